// HausdorffDistanceLoss_85761906966840
// MI455X (gfx1250) — compile-verified
//
#include <hip/hip_runtime.h>
#include <hip/hip_bf16.h>

typedef float v2f __attribute__((ext_vector_type(2)));
typedef float v8f __attribute__((ext_vector_type(8)));

#define HH 256
#define WW 256
#define HW 65536
#define BIG 512          // H + W
#define NSLICE 16        // 8 gt slices + 8 pred slices
#define INV_COUNT (0.5f / 524288.0f)   // 0.5 / (4*2*256*256)

// ds_swizzle_b32, group-of-32 XOR mode: offset = {xor[4:0], or=0, and=0x1f}
// Pattern must be an integer constant expression -> template parameter.
template <int PAT>
__device__ __forceinline__ float swz_xor_add(float x) {
  int y = __builtin_amdgcn_ds_swizzle(__float_as_int(x), PAT);
  return x + __int_as_float(y);
}

// ---------------------------------------------------------------------------
// Stage 1: per-column 1D distance (exact), then square. One thread per column,
// one block per slice. All global accesses are coalesced (thread j <-> col j).
// sqbuf[s][i][j] = dcol^2 as float (exact integer <= 262144).
// ---------------------------------------------------------------------------
__global__ __launch_bounds__(256) void edt_cols_kernel(
    const float* __restrict__ pred, const int* __restrict__ target,
    float* __restrict__ sqbuf) {
  const int s = blockIdx.x;          // slice 0..15
  const int j = threadIdx.x;         // column
  const int side = s >> 3;           // 0 = gt sites, 1 = pred sites
  const int r = s & 7;
  const int n = r >> 1;
  const int c = (r & 1) + 1;         // class / channel 1..2
  float* out = sqbuf + s * HW;

  const int*   tg = target + (n << 16);
  const float* pr = pred + ((n * 3 + c) * HH) * WW;

  // forward pass: distance to nearest site above (store temporarily)
  int last = -BIG;
  for (int i = 0; i < HH; ++i) {
    bool site = (side == 0) ? (tg[(i << 8) + j] == c)
                            : (pr[(i << 8) + j] >= 0.5f);
    if (site) last = i;
    out[(i << 8) + j] = (float)(i - last);      // <= 255 + 512, exact
  }
  // backward pass: combine with distance to nearest site below, cap, square
  int nxt = HH + BIG;
  for (int i = HH - 1; i >= 0; --i) {
    bool site = (side == 0) ? (tg[(i << 8) + j] == c)
                            : (pr[(i << 8) + j] >= 0.5f);
    if (site) nxt = i;
    int down = (int)out[(i << 8) + j];
    int d = min(min(down, nxt - i), BIG);
    out[(i << 8) + j] = (float)(d * d);
  }
}

// ---------------------------------------------------------------------------
// Stage 2: row-wise exact min-plus pass  d2[j] = min_k( sq[k] + (j-k)^2 ),
// fused with the weighted accumulation  acc += w[j] * d2[j].
// Block = (slice, 16-row band); thread = output column j. The whole band
// (16 KB) is staged into LDS behind a SINGLE barrier, then the 8 waves run
// the min-plus rows completely barrier-free. Inner loop: float4 LDS
// broadcast reads (ds_load_b128) feeding 4 independent fma/min chains.
// Epilogue: exact f32 wave reduction on the matrix pipe via
// V_WMMA_F32_16X16X4_F32 with an all-ones A matrix (D[m,n] = sum_k B[k,n]
// => summing one D row gives the sum of all 64 B slots, independent of the
// B VGPR layout), then ds_swizzle XOR folds the 16 column sums.
// ---------------------------------------------------------------------------
__global__ __launch_bounds__(256) void rowpass_kernel(
    const float* __restrict__ pred, const int* __restrict__ target,
    const float* __restrict__ sqbuf, float* __restrict__ partials) {
  const int blk = blockIdx.x;        // 0..255
  const int s = blk >> 4;            // slice
  const int band = blk & 15;         // 16-row band within slice
  const int j = threadIdx.x;
  const int side = s >> 3;
  const int r = s & 7;
  const int n = r >> 1;
  const int c = (r & 1) + 1;

  const float* sq = sqbuf + s * HW;
  const int*   tg = target + (n << 16);
  const float* pr = pred + ((n * 3 + c) * HH) * WW;

  __shared__ float srows[16 * WW];   // 16 KB: the block's entire row band
  __shared__ float wavesum[8];

  // stage the whole band with one barrier (coalesced: thread j <-> col j)
#pragma unroll
  for (int ri = 0; ri < 16; ++ri) {
    srows[ri * WW + j] = sq[(((band << 4) + ri) << 8) + j];
  }
  __syncthreads();

  const float jf = (float)j;
  float acc = 0.0f;

  for (int ri = 0; ri < 16; ++ri) {
    const int i = (band << 4) + ri;
    // weight: term1 uses pred_fg against gt-EDT, term2 uses fg against pred-EDT
    float w = (side == 0) ? pr[(i << 8) + j]
                          : ((tg[(i << 8) + j] == c) ? 1.0f : 0.0f);
    if (ri < 15) {
      // pull the next weight row toward the WGP while we grind min-plus
      __builtin_prefetch((side == 0) ? (const void*)&pr[((i + 1) << 8) + j]
                                     : (const void*)&tg[((i + 1) << 8) + j],
                         0, 1);
    }

    const float4* row4 = reinterpret_cast<const float4*>(&srows[ri * WW]);
    // 4 independent fma/min chains (quarters the critical-path latency)
    float b0 = 3.4e38f, b1 = 3.4e38f, b2 = 3.4e38f, b3 = 3.4e38f;
    float d0 = jf, d1 = jf - 1.0f, d2 = jf - 2.0f, d3 = jf - 3.0f;
#pragma unroll 4
    for (int k4 = 0; k4 < WW / 4; ++k4) {
      float4 q = row4[k4];                      // LDS broadcast b128
      b0 = fminf(b0, fmaf(d0, d0, q.x));  d0 -= 4.0f;
      b1 = fminf(b1, fmaf(d1, d1, q.y));  d1 -= 4.0f;
      b2 = fminf(b2, fmaf(d2, d2, q.z));  d2 -= 4.0f;
      b3 = fminf(b3, fmaf(d3, d3, q.w));  d3 -= 4.0f;
    }
    float best = fminf(fminf(b0, b1), fminf(b2, b3));
    acc = fmaf(w, best, acc);
  }

  // ---- per-wave exact f32 reduction via WMMA (EXEC all-ones here) ----
  v2f a; a[0] = 1.0f; a[1] = 1.0f;             // A == all ones (layout agnostic)
  v2f b; b[0] = acc;  b[1] = 0.0f;             // one accumulator per lane
  v8f cacc = {};
  cacc = __builtin_amdgcn_wmma_f32_16x16x4_f32(
      /*neg_a=*/false, a, /*neg_b=*/false, b,
      /*c_mod=*/(short)0, cacc, /*reuse_a=*/false, /*reuse_b=*/false);
  // D rows are identical; C VGPR0 lane L holds colsum[L % 16]. XOR-fold the
  // 16-lane groups with ds_swizzle (both halves yield the same wave total).
  float rsum = cacc[0];
  rsum = swz_xor_add<(1 << 10) | 0x1f>(rsum);  // SWAPX1
  rsum = swz_xor_add<(2 << 10) | 0x1f>(rsum);  // SWAPX2
  rsum = swz_xor_add<(4 << 10) | 0x1f>(rsum);  // SWAPX4
  rsum = swz_xor_add<(8 << 10) | 0x1f>(rsum);  // SWAPX8

  const int lane = j & 31;
  const int wave = j >> 5;
  if (lane == 0) wavesum[wave] = rsum;
  __syncthreads();
  if (j == 0) {
    float t = 0.0f;
#pragma unroll
    for (int w8 = 0; w8 < 8; ++w8) t += wavesum[w8];
    partials[blk] = t;
  }
}

// ---------------------------------------------------------------------------
// Stage 3: deterministic fixed-order reduction of the 256 block partials.
// ---------------------------------------------------------------------------
__global__ __launch_bounds__(256) void finalize_kernel(
    const float* __restrict__ partials, float* __restrict__ out) {
  __shared__ float buf[256];
  const int j = threadIdx.x;
  buf[j] = partials[j];
  __syncthreads();
  for (int st = 128; st > 0; st >>= 1) {
    if (j < st) buf[j] += buf[j + st];
    __syncthreads();
  }
  if (j == 0) out[0] = buf[0] * INV_COUNT;
}

// ---------------------------------------------------------------------------
extern "C" void kernel_launch(void* const* d_in, const int* in_sizes, int n_in,
                              void* d_out, int out_size, void* d_ws, size_t ws_size,
                              hipStream_t stream) {
  (void)in_sizes; (void)n_in; (void)out_size; (void)ws_size;
  const float* pred   = (const float*)d_in[0];   // [4,3,256,256] f32
  const int*   target = (const int*)d_in[1];     // [4,256,256] int

  float* sqbuf    = (float*)d_ws;                // 16 * 65536 floats = 4 MB
  float* partials = sqbuf + NSLICE * HW;         // 256 floats

  edt_cols_kernel<<<NSLICE, 256, 0, stream>>>(pred, target, sqbuf);
  rowpass_kernel<<<NSLICE * 16, 256, 0, stream>>>(pred, target, sqbuf, partials);
  finalize_kernel<<<1, 256, 0, stream>>>(partials, (float*)d_out);
}